// CrossLayer_20392504721652
// MI455X (gfx1250) — compile-verified
//
#include <hip/hip_runtime.h>
#include <hip/hip_bf16.h>

// Problem constants (from reference): B=16384 rows, D=1024 features, L=8 layers.
#define BROWS 16384
#define DDIM  1024
#define LLAY  8
#define WSTRIDE 1026   // LDS row stride in dwords: 1026*4=4104B (8B-aligned rows),
                       // 1026 % 64 == 2 -> at worst 2-way bank conflict on b64 reads

typedef float v2f __attribute__((ext_vector_type(2)));
typedef float v8f __attribute__((ext_vector_type(8)));

// ---------------------------------------------------------------------------
// Kernel 1: v[i] = dot(b[i-1], W[i]) for i=1..7, v[0]=0.  One block, 8 waves.
// ---------------------------------------------------------------------------
__global__ void cross_compute_v(const float* __restrict__ W,
                                const float* __restrict__ b,
                                float* __restrict__ v) {
    const int wave = threadIdx.x >> 5;
    const int lane = threadIdx.x & 31;
    float s = 0.0f;
    if (wave > 0) {
        const float* br = b + (size_t)(wave - 1) * DDIM;
        const float* wr = W + (size_t)wave * DDIM;
        #pragma unroll 8
        for (int k = lane; k < DDIM; k += 32) s += br[k] * wr[k];
    }
    #pragma unroll
    for (int off = 16; off > 0; off >>= 1) s += __shfl_xor(s, off, 32);
    if (lane == 0) v[wave] = s;
}

// ---------------------------------------------------------------------------
// Kernel 2: U = x @ W^T via V_WMMA_F32_16X16X4_F32 (one wave per 16-row tile),
// then the 8-step alpha recurrence in-wave via an LDS transpose.
//   A frag (16x4 f32): lane<16 -> M=lane,   K = 0,1   (VGPR0,1)
//                      lane>=16-> M=lane-16,K = 2,3
//   B frag (4x16 f32): lane<16 -> N=lane,   rows K=0,1; lane>=16 -> N=lane-16, K=2,3
//   W is staged in LDS zero-padded to 16 rows, so the inner loop is branch-free:
//   global_load_b64 (A) + ds_load_b64 (B) + v_wmma per K-step.
// ---------------------------------------------------------------------------
__global__ void cross_compute_alpha(const float* __restrict__ x,
                                    const float* __restrict__ W,
                                    const float* __restrict__ v,
                                    float* __restrict__ alpha) {
    __shared__ float ldsW[16 * WSTRIDE];   // 65,664 B: rows 0..7 = W, rows 8..15 = 0
    __shared__ float Ut[8][16][17];        // per-wave transpose buffer

    // ---- Stage zero-padded W into LDS (once per block; W is L2-hot) ----
    for (int t = threadIdx.x; t < 16 * (DDIM / 4); t += blockDim.x) {
        const int row = t >> 8;          // / (DDIM/4)
        const int j4  = t & 255;
        float2 lo = make_float2(0.f, 0.f), hi = make_float2(0.f, 0.f);
        if (row < LLAY) {
            const float4 w4 = ((const float4*)(W + (size_t)row * DDIM))[j4];
            lo = make_float2(w4.x, w4.y);
            hi = make_float2(w4.z, w4.w);
        }
        float* dst = &ldsW[row * WSTRIDE + (j4 << 2)];
        *(float2*)(dst)     = lo;
        *(float2*)(dst + 2) = hi;
    }
    __syncthreads();

    const int waveInBlock = threadIdx.x >> 5;
    const int lane        = threadIdx.x & 31;
    const int waveGlobal  = blockIdx.x * (blockDim.x >> 5) + waveInBlock;
    const int row0        = waveGlobal * 16;

    const int m  = lane & 15;          // M (for A) / N (for B)
    const int kk = (lane >> 4) << 1;   // 0 or 2: which K-pair this half-wave holds

    const float* arow = x + (size_t)(row0 + m) * DDIM + kk;
    const float* lw   = &ldsW[m * WSTRIDE + kk];

    v8f c = {0.f, 0.f, 0.f, 0.f, 0.f, 0.f, 0.f, 0.f};

    #pragma unroll 8
    for (int k = 0; k < DDIM; k += 4) {
        const float2 af = *(const float2*)(arow + k);   // global_load_b64
        const float2 bf = *(const float2*)(lw + k);     // ds_load_b64 (zero-padded N>=8)

        v2f av; av.x = af.x; av.y = af.y;
        v2f bv; bv.x = bf.x; bv.y = bf.y;
        // D = A(16x4) * B(4x16) + C   -> v_wmma_f32_16x16x4_f32
        c = __builtin_amdgcn_wmma_f32_16x16x4_f32(
                /*neg_a=*/false, av, /*neg_b=*/false, bv,
                /*c_mod=*/(short)0, c, /*reuse_a=*/false, /*reuse_b=*/false);
    }

    // Scatter D tile into LDS: Ut[M][N]
    #pragma unroll
    for (int r = 0; r < 8; ++r) {
        const int M = r + ((lane >> 4) << 3);
        Ut[waveInBlock][M][m] = c[r];
    }
    __asm__ volatile("s_wait_dscnt 0x0" ::: "memory");

    // alpha recurrence: alpha_{i+1} = alpha_i * U[r][i] + v_i + 1
    if (lane < 16) {
        float al = 1.0f;
        #pragma unroll
        for (int i = 0; i < LLAY; ++i)
            al = al * Ut[waveInBlock][lane][i] + v[i] + 1.0f;
        alpha[row0 + lane] = al;
    }
}

// ---------------------------------------------------------------------------
// Kernel 3: out = alpha[row] * x + b[7]   (float4 streaming; sets the runtime:
// 64MB x read (L2-resident from kernel 2) + 64MB out write ~= 5.5us at 23.3TB/s)
// ---------------------------------------------------------------------------
__global__ void cross_finalize(const float* __restrict__ x,
                               const float* __restrict__ blast,
                               const float* __restrict__ alpha,
                               float* __restrict__ out) {
    const int cols4 = DDIM / 4;
    const size_t gid = (size_t)blockIdx.x * blockDim.x + threadIdx.x;  // over B*D/4
    const int row = (int)(gid / cols4);
    const int c4  = (int)(gid % cols4);

    const float  a  = alpha[row];
    const float4 x4 = ((const float4*)x)[gid];
    const float4 b4 = ((const float4*)blast)[c4];

    float4 o;
    o.x = fmaf(a, x4.x, b4.x);
    o.y = fmaf(a, x4.y, b4.y);
    o.z = fmaf(a, x4.z, b4.z);
    o.w = fmaf(a, x4.w, b4.w);
    ((float4*)out)[gid] = o;
}

// ---------------------------------------------------------------------------
extern "C" void kernel_launch(void* const* d_in, const int* in_sizes, int n_in,
                              void* d_out, int out_size, void* d_ws, size_t ws_size,
                              hipStream_t stream) {
    (void)in_sizes; (void)n_in; (void)out_size; (void)ws_size;

    const float* x = (const float*)d_in[0];   // [B, D]
    const float* W = (const float*)d_in[1];   // [L, D]
    const float* b = (const float*)d_in[2];   // [L, D]
    float* out = (float*)d_out;               // [B, D]

    float* v     = (float*)d_ws;              // 8 floats
    float* alpha = v + 8;                     // B floats  (total ws use ~64KB)

    // 1) per-layer scalars v_i = b_{i-1} . w_i
    cross_compute_v<<<1, 256, 0, stream>>>(W, b, v);

    // 2) U = x @ W^T via WMMA + alpha recurrence. One wave per 16 rows.
    //    B/16 = 1024 waves -> 128 blocks of 256 threads (8 waves).
    cross_compute_alpha<<<BROWS / 16 / 8, 256, 0, stream>>>(x, W, v, alpha);

    // 3) out = alpha * x + b[L-1]
    const float* blast = b + (size_t)(LLAY - 1) * DDIM;
    cross_finalize<<<(BROWS * (DDIM / 4)) / 256, 256, 0, stream>>>(x, blast, alpha, out);
}